// SimpleGCNNet_36859409334535
// MI455X (gfx1250) — compile-verified
//
#include <hip/hip_runtime.h>

typedef __attribute__((ext_vector_type(2))) float v2f;
typedef __attribute__((ext_vector_type(8))) float v8f;

#define D_FEAT 512
#define N_OUT  64
// K-pair rows: p = k/2 in [0,256). Row stride 80 float2 (160 dwords):
// 160 mod 64 = 32 -> half-wave lane groups land on disjoint 32-bank halves,
// so a wave32 ds_load_b64 uses all 64 LDS banks conflict-free.
#define PAIR_ROWS   (D_FEAT / 2)
#define PAIR_STRIDE 80

// ---------------------------------------------------------------------------
// Kernel 1: h = x @ W via V_WMMA_F32_16X16X4_F32.
// Block = 256 threads (8 waves). Each wave owns a 16-row tile, all 64 cols.
// W staged in LDS K-pair-interleaved: P[p][n] = (W[2p][n], W[2p+1][n]),
// so each B fragment is ONE contiguous ds_load_b64 into an aligned VGPR pair.
// ---------------------------------------------------------------------------
__global__ __launch_bounds__(256) void gcn_gemm_wmma(
    const float* __restrict__ x, const float* __restrict__ W,
    float* __restrict__ h, int n_nodes)
{
    __shared__ float2 Wp[PAIR_ROWS * PAIR_STRIDE];   // 160 KB

    // Cooperative pair-interleaved stage of W.
    // 256 p-rows x 16 float4-chunks = 4096 chunks; 16 per thread.
    for (int i = threadIdx.x; i < PAIR_ROWS * 16; i += 256) {
        const int p  = i >> 4;            // K-pair index
        const int n4 = (i & 15) << 2;     // column chunk base (n..n+3)
        const float4 r0 = *reinterpret_cast<const float4*>(W + (2 * p + 0) * N_OUT + n4);
        const float4 r1 = *reinterpret_cast<const float4*>(W + (2 * p + 1) * N_OUT + n4);
        float4* dst = reinterpret_cast<float4*>(&Wp[p * PAIR_STRIDE + n4]);
        float4 lo, hi;
        lo.x = r0.x; lo.y = r1.x; lo.z = r0.y; lo.w = r1.y;   // (P[n4], P[n4+1])
        hi.x = r0.z; hi.y = r1.z; hi.z = r0.w; hi.w = r1.w;   // (P[n4+2], P[n4+3])
        dst[0] = lo;
        dst[1] = hi;
    }
    __syncthreads();

    const int lane = threadIdx.x & 31;
    const int wave = threadIdx.x >> 5;
    const int l16  = lane & 15;
    const int g    = lane >> 4;                       // half-wave group 0/1
    const long rowBase = (long)blockIdx.x * 128 + (long)wave * 16;
    if (rowBase >= n_nodes) return;                   // wave-uniform exit

    // A fragment: lane reads x[rowBase + l16][k + 2g + {0,1}] (one b64 load)
    const float* aptr = x + (rowBase + l16) * (long)D_FEAT + 2 * g;
    // B fragment: lane reads Wp[(k/2 + g)*PAIR_STRIDE + n0 + l16] (one b64 load)
    const float2* bptr = &Wp[(long)g * PAIR_STRIDE + l16];

    v8f acc0 = {}, acc1 = {}, acc2 = {}, acc3 = {};

    for (int k = 0; k < D_FEAT; k += 4) {
        const v2f a = *reinterpret_cast<const v2f*>(aptr + k);

        const float2* wb = bptr + (k >> 1) * PAIR_STRIDE;
        const v2f b0 = *reinterpret_cast<const v2f*>(wb +  0);
        const v2f b1 = *reinterpret_cast<const v2f*>(wb + 16);
        const v2f b2 = *reinterpret_cast<const v2f*>(wb + 32);
        const v2f b3 = *reinterpret_cast<const v2f*>(wb + 48);

        acc0 = __builtin_amdgcn_wmma_f32_16x16x4_f32(false, a, false, b0, (short)0, acc0, false, false);
        acc1 = __builtin_amdgcn_wmma_f32_16x16x4_f32(false, a, false, b1, (short)0, acc1, false, false);
        acc2 = __builtin_amdgcn_wmma_f32_16x16x4_f32(false, a, false, b2, (short)0, acc2, false, false);
        acc3 = __builtin_amdgcn_wmma_f32_16x16x4_f32(false, a, false, b3, (short)0, acc3, false, false);
    }

    // C/D layout: VGPR v -> M = v + 8g, N = l16 (+16*ct)
    float* hb = h + rowBase * N_OUT + l16;
#pragma unroll
    for (int v = 0; v < 8; ++v) {
        const long m = v + 8 * g;
        hb[m * N_OUT +  0] = acc0[v];
        hb[m * N_OUT + 16] = acc1[v];
        hb[m * N_OUT + 32] = acc2[v];
        hb[m * N_OUT + 48] = acc3[v];
    }
}

// ---------------------------------------------------------------------------
// Kernel 2: zero the aggregation buffer (d_out) with float4 stores.
// ---------------------------------------------------------------------------
__global__ __launch_bounds__(256) void gcn_zero(float4* __restrict__ p, long n4)
{
    long i = (long)blockIdx.x * blockDim.x + threadIdx.x;
    const long stride = (long)gridDim.x * blockDim.x;
    const float4 z = {0.f, 0.f, 0.f, 0.f};
    for (; i < n4; i += stride) p[i] = z;
}

// ---------------------------------------------------------------------------
// Kernel 3: edge scatter. One wave per edge; lane owns columns {2*lane, 2*lane+1}.
// agg[dst] += h[src] * w  via hardware f32 atomics (L2 atomic units).
// h (25.6 MB) is L2-resident, so gathers mostly hit L2.
// ---------------------------------------------------------------------------
__global__ __launch_bounds__(256) void gcn_scatter(
    const float* __restrict__ h, const int* __restrict__ esrc,
    const int* __restrict__ edst, const float* __restrict__ ew,
    float* __restrict__ agg, int n_edges)
{
    const int wave_id = (int)(((long)blockIdx.x * blockDim.x + threadIdx.x) >> 5);
    const int lane = threadIdx.x & 31;
    if (wave_id >= n_edges) return;

    const int   s  = esrc[wave_id];
    const int   d  = edst[wave_id];
    const float wt = ew[wave_id];

    const float2 hv = reinterpret_cast<const float2*>(h + (long)s * N_OUT)[lane];
    float* ap = agg + (long)d * N_OUT + 2 * lane;
#if defined(__HIP_DEVICE_COMPILE__)
    unsafeAtomicAdd(ap + 0, hv.x * wt);
    unsafeAtomicAdd(ap + 1, hv.y * wt);
#endif
}

// ---------------------------------------------------------------------------
// Kernel 4: row softmax. One wave per 64-wide row; 2 elements per lane;
// wave32 shuffle reductions.
// ---------------------------------------------------------------------------
__global__ __launch_bounds__(256) void gcn_softmax(float* __restrict__ agg, int n_nodes)
{
    const int wave_id = (int)(((long)blockIdx.x * blockDim.x + threadIdx.x) >> 5);
    const int lane = threadIdx.x & 31;
    if (wave_id >= n_nodes) return;

    float2* row = reinterpret_cast<float2*>(agg + (long)wave_id * N_OUT);
    const float2 v = row[lane];

    float m = fmaxf(v.x, v.y);
#pragma unroll
    for (int off = 16; off > 0; off >>= 1)
        m = fmaxf(m, __shfl_xor(m, off, 32));

    const float e0 = __expf(v.x - m);
    const float e1 = __expf(v.y - m);
    float s = e0 + e1;
#pragma unroll
    for (int off = 16; off > 0; off >>= 1)
        s += __shfl_xor(s, off, 32);

    const float inv = 1.0f / s;
    float2 o; o.x = e0 * inv; o.y = e1 * inv;
    row[lane] = o;
}

// ---------------------------------------------------------------------------
extern "C" void kernel_launch(void* const* d_in, const int* in_sizes, int n_in,
                              void* d_out, int out_size, void* d_ws, size_t ws_size,
                              hipStream_t stream)
{
    const float* x    = (const float*)d_in[0];
    const int*   esrc = (const int*)d_in[1];
    const int*   edst = (const int*)d_in[2];
    const float* ew   = (const float*)d_in[3];
    const float* W    = (const float*)d_in[4];

    const int n_nodes = in_sizes[0] / D_FEAT;   // 100000
    const int n_edges = in_sizes[1];            // 1600000

    float* agg = (float*)d_out;                 // [n_nodes, 64] accum + final output
    float* h   = (float*)d_ws;                  // [n_nodes, 64] projection scratch

    // 1) zero agg
    {
        const long n4 = ((long)n_nodes * N_OUT) / 4;
        const int blocks = (int)((n4 + 255) / 256);
        gcn_zero<<<blocks, 256, 0, stream>>>((float4*)agg, n4);
    }
    // 2) h = x @ W (WMMA)
    {
        const int blocks = (n_nodes + 127) / 128;   // 128 rows per block (8 waves x 16)
        gcn_gemm_wmma<<<blocks, 256, 0, stream>>>(x, W, h, n_nodes);
    }
    // 3) scatter-add over edges (1 wave per edge)
    {
        const int blocks = (int)(((long)n_edges * 32 + 255) / 256);
        gcn_scatter<<<blocks, 256, 0, stream>>>(h, esrc, edst, ew, agg, n_edges);
    }
    // 4) softmax rows (1 wave per row)
    {
        const int blocks = (int)(((long)n_nodes * 32 + 255) / 256);
        gcn_softmax<<<blocks, 256, 0, stream>>>(agg, n_nodes);
    }
}